// EmaLayerNorm_41094247088403
// MI455X (gfx1250) — compile-verified
//
#include <hip/hip_runtime.h>
#include <hip/hip_bf16.h>

// ---------------------------------------------------------------------------
// EmaLayerNorm for MI455X (gfx1250, wave32).
// Shapes fixed by reference: B=8, S=8192, D=512, f32 in/out.
// ---------------------------------------------------------------------------

#define B_DIM 8
#define S_DIM 8192
#define D_DIM 512
#define N_ROWS (B_DIM * S_DIM)          // 65536
#define Y_ELEMS ((size_t)N_ROWS * D_DIM) // 33554432
#define EPS_VAL 1e-12f

typedef __attribute__((ext_vector_type(2))) float v2f;
typedef __attribute__((ext_vector_type(8))) float v8f;

__device__ __forceinline__ float v8_get(v8f v, int i) {
    switch (i & 7) {
        case 0: return v[0];
        case 1: return v[1];
        case 2: return v[2];
        case 3: return v[3];
        case 4: return v[4];
        case 5: return v[5];
        case 6: return v[6];
        default: return v[7];
    }
}

// ---------------------------------------------------------------------------
// Pass 1: per-row mean / std via WMMA.
// One wave handles a tile of 16 rows x 512 cols.
//   acc_S += A x ones   -> row sums (all 16 columns identical)
//   acc_G += A x A      -> Gram matrix; diagonal = row sum-of-squares
// A-tile f32 16x16x4 layout == B-tile layout of its transpose, so the same
// two-VGPR register pair feeds both operands.
// ---------------------------------------------------------------------------
__global__ __launch_bounds__(256)
void ema_stats_kernel(const float* __restrict__ x,
                      float* __restrict__ mu,
                      float* __restrict__ sig) {
    const int wave_in_blk = threadIdx.x >> 5;
    const int lane        = threadIdx.x & 31;
    const int tile        = blockIdx.x * 8 + wave_in_blk;   // 0..4095

    const int m    = lane & 15;            // row within tile (per A layout)
    const int koff = (lane >> 4) * 2;      // K sub-offset for hi half-wave

    const size_t row0 = (size_t)tile * 16;
    const float* rowp = x + (row0 + (size_t)m) * D_DIM + koff;

    v8f accS = {};   // sums
    v8f accG = {};   // Gram (diag = sum of squares)
    const v2f ones = {1.0f, 1.0f};

#pragma unroll 8
    for (int k = 0; k < D_DIM; k += 4) {
        v2f a = *(const v2f*)(rowp + k);
        // (neg_a, A, neg_b, B, c_mod, C, reuse_a, reuse_b)
        accS = __builtin_amdgcn_wmma_f32_16x16x4_f32(
            false, a, false, ones, (short)0, accS, false, false);
        accG = __builtin_amdgcn_wmma_f32_16x16x4_f32(
            false, a, false, a, (short)0, accG, false, false);
    }

    // C layout: VGPR r, lanes 0-15 -> D[M=r][N=lane]; lanes 16-31 -> D[M=r+8][N=lane-16].
    // Row sums (columns identical): lanes 0-15 expose rows 0-7 via index,
    // lanes 16-31 expose rows 8-15.
    float vS = v8_get(accS, lane & 7);
    // Gram diagonal D[M][M]: valid on lanes 0-7 (rows 0-7) and 24-31 (rows 8-15).
    float vG = v8_get(accG, lane & 7);

    const int srcS = (m < 8) ? m : (m + 8);    // lanes 0-7 / 16-23
    const int srcG = (m < 8) ? m : (m + 16);   // lanes 0-7 / 24-31
    float rsum = __shfl(vS, srcS, 32);
    float rssq = __shfl(vG, srcG, 32);

    if (lane < 16) {
        float mean = rsum * (1.0f / (float)D_DIM);
        float var  = (rssq - (float)D_DIM * mean * mean) * (1.0f / (float)(D_DIM - 1));
        float sd   = sqrtf(fmaxf(var, 0.0f));
        mu[row0 + m]  = mean;
        sig[row0 + m] = sd;
    }
}

// ---------------------------------------------------------------------------
// Pass 2: per-batch weighted prefix sum over S.
//   mu_cum[s]    = alpha*h_mu  + cumsum((1-a)*a^s * mu[s])
//   sigma_cum[s] = max(alpha*h_sig + cumsum((1-a)*a^s * sig[s]), EPS)
// 1024 threads, 8 elements each; Hillis-Steele block scan over thread totals.
// ---------------------------------------------------------------------------
__global__ __launch_bounds__(1024)
void ema_scan_kernel(const float* __restrict__ mu,
                     const float* __restrict__ sig,
                     float* __restrict__ mu_cum,
                     float* __restrict__ sig_cum,
                     const float* __restrict__ h,
                     const float* __restrict__ alpha_logit,
                     float* __restrict__ h_out) {
    const int b = blockIdx.x;
    const int t = threadIdx.x;            // 0..1023
    const int PER = S_DIM / 1024;         // 8
    const int base = t * PER;
    const size_t roff = (size_t)b * S_DIM;

    const float alpha = 1.0f / (1.0f + expf(-alpha_logit[0]));
    const float one_m_a = 1.0f - alpha;
    const float l2a = log2f(alpha);

    float cmu[8], csg[8];
#pragma unroll
    for (int i = 0; i < PER; ++i) {
        int s = base + i;
        float w = one_m_a * exp2f(l2a * (float)s);   // (1-a)*a^s
        cmu[i] = w * mu[roff + s];
        csg[i] = w * sig[roff + s];
    }
    // local inclusive scan
#pragma unroll
    for (int i = 1; i < PER; ++i) { cmu[i] += cmu[i - 1]; csg[i] += csg[i - 1]; }

    __shared__ float smu[2][1024];
    __shared__ float ssg[2][1024];
    smu[0][t] = cmu[PER - 1];
    ssg[0][t] = csg[PER - 1];
    __syncthreads();

    int src = 0;
    for (int off = 1; off < 1024; off <<= 1) {
        float vm = smu[src][t];
        float vs = ssg[src][t];
        if (t >= off) { vm += smu[src][t - off]; vs += ssg[src][t - off]; }
        smu[src ^ 1][t] = vm;
        ssg[src ^ 1][t] = vs;
        __syncthreads();
        src ^= 1;
    }

    float excl_mu = (t > 0) ? smu[src][t - 1] : 0.0f;
    float excl_sg = (t > 0) ? ssg[src][t - 1] : 0.0f;

    const float hmu = h[b * 2 + 0] * alpha;
    const float hsg = h[b * 2 + 1] * alpha;

#pragma unroll
    for (int i = 0; i < PER; ++i) {
        int s = base + i;
        float m = hmu + excl_mu + cmu[i];
        float g = hsg + excl_sg + csg[i];
        g = fmaxf(g, EPS_VAL);
        mu_cum[roff + s]  = m;
        sig_cum[roff + s] = g;
        if (s == S_DIM - 1) {            // h_new = [mu_cum[-1], sigma_cum[-1]]
            h_out[b * 2 + 0] = m;
            h_out[b * 2 + 1] = g;
        }
    }
}

// ---------------------------------------------------------------------------
// Pass 3: streaming normalize, float4-vectorized (B128 loads/stores).
// x is L2-resident (128 MiB < 192 MB) after pass 1, so HBM traffic here is
// essentially the y write stream.
// ---------------------------------------------------------------------------
__global__ __launch_bounds__(256)
void ema_norm_kernel(const float* __restrict__ x,
                     const float* __restrict__ mu_cum,
                     const float* __restrict__ sig_cum,
                     const float* __restrict__ gamma,
                     const float* __restrict__ beta,
                     float* __restrict__ y) {
    const size_t i = (size_t)blockIdx.x * blockDim.x + threadIdx.x; // float4 idx
    const int d4  = (int)(i & (D_DIM / 4 - 1));   // 0..127
    const size_t row = i >> 7;                    // /128

    float4 xv = ((const float4*)x)[i];
    float m = mu_cum[row];
    float inv = 1.0f / sig_cum[row];
    float4 g  = ((const float4*)gamma)[d4];
    float4 bt = ((const float4*)beta)[d4];

    float4 r;
    r.x = (xv.x - m) * inv * g.x + bt.x;
    r.y = (xv.y - m) * inv * g.y + bt.y;
    r.z = (xv.z - m) * inv * g.z + bt.z;
    r.w = (xv.w - m) * inv * g.w + bt.w;
    ((float4*)y)[i] = r;
}

// ---------------------------------------------------------------------------
extern "C" void kernel_launch(void* const* d_in, const int* in_sizes, int n_in,
                              void* d_out, int out_size, void* d_ws, size_t ws_size,
                              hipStream_t stream) {
    const float* x           = (const float*)d_in[0];
    const float* h           = (const float*)d_in[1];
    const float* gamma       = (const float*)d_in[2];
    const float* beta        = (const float*)d_in[3];
    const float* alpha_logit = (const float*)d_in[4];

    float* y     = (float*)d_out;
    float* h_out = (float*)d_out + Y_ELEMS;

    float* wsf      = (float*)d_ws;
    float* mu       = wsf;                 // 65536
    float* sig      = wsf + N_ROWS;        // 65536
    float* mu_cum   = wsf + 2 * N_ROWS;    // 65536
    float* sig_cum  = wsf + 3 * N_ROWS;    // 65536  -> 1 MiB total

    // Pass 1: 4096 tiles of 16 rows, 8 waves (256 thr) per block.
    ema_stats_kernel<<<N_ROWS / 16 / 8, 256, 0, stream>>>(x, mu, sig);

    // Pass 2: one block per batch.
    ema_scan_kernel<<<B_DIM, 1024, 0, stream>>>(mu, sig, mu_cum, sig_cum,
                                                h, alpha_logit, h_out);

    // Pass 3: 8M float4 elements.
    ema_norm_kernel<<<(unsigned)(Y_ELEMS / 4 / 256), 256, 0, stream>>>(
        x, mu_cum, sig_cum, gamma, beta, y);
}